// VisionEncoder_83777632076510
// MI455X (gfx1250) — compile-verified
//
#include <hip/hip_runtime.h>
#include <math.h>

// MoE VisionEncoder for gfx1250: top-2 sparse expert evaluation.
// Pointwise convs = per-(sample,expert) GEMMs on V_WMMA_F32_16X16X4_F32 (fp32 precise),
// templated on K (fully unrolled, branch-free) with MT accumulator tiles per wave.
// GroupNorm folded into per-channel affine; depthwise conv scalar VALU; gate-weighted
// accumulation via 2-way-commutative float atomics (deterministic).

typedef __attribute__((ext_vector_type(2))) float v2f;
typedef __attribute__((ext_vector_type(8))) float v8f;

#define NEXP   8
#define BATCH  64
#define JOBS   128   // BATCH * top-2
#define CHUNK  32    // jobs processed per kernel wave-set
#define MAXCH  256   // max channels (stage 4 expand)

// ---------------- gating: logits -> softmax -> top2 -> renormalized gates -------------
__global__ void k_gating(const float* __restrict__ gi,
                         const float* __restrict__ wg0, const float* __restrict__ wg1,
                         const float* __restrict__ wg2, const float* __restrict__ wg3,
                         const float* __restrict__ wg4,
                         const int* __restrict__ task_p,
                         int* __restrict__ jobs_e, float* __restrict__ jobs_g)
{
    int b = threadIdx.x;
    if (b >= BATCH) return;
    int task = task_p[0];
    const float* wgs[5] = {wg0, wg1, wg2, wg3, wg4};
    for (int s = 0; s < 5; ++s) {
        const float* w = wgs[s] + (size_t)task * 16 * NEXP;   // (16, 8) slice
        float logits[NEXP];
        for (int e = 0; e < NEXP; ++e) {
            float acc = 0.f;
            for (int d = 0; d < 16; ++d) acc += gi[b * 16 + d] * w[d * NEXP + e];
            logits[e] = acc;
        }
        int i1 = 0;
        for (int e = 1; e < NEXP; ++e) if (logits[e] > logits[i1]) i1 = e;
        int i2 = (i1 == 0) ? 1 : 0;
        for (int e = 0; e < NEXP; ++e) if (e != i1 && logits[e] > logits[i2]) i2 = e;
        // renormalized top-2 softmax gates = e^l1/(e^l1+e^l2), e^l2/(...)
        float m  = logits[i1];
        float e1 = expf(logits[i1] - m), e2 = expf(logits[i2] - m);
        float inv = 1.f / (e1 + e2);
        jobs_e[s * JOBS + b * 2 + 0] = i1; jobs_g[s * JOBS + b * 2 + 0] = e1 * inv;
        jobs_e[s * JOBS + b * 2 + 1] = i2; jobs_g[s * JOBS + b * 2 + 1] = e2 * inv;
    }
}

// ---------------- pointwise conv as GEMM on fp32 WMMA ---------------------------------
// D(M,N) = W[e](M,K) * X(K,N).  One wave computes MT vertically-stacked 16x16 tiles,
// sharing the B (activation) fragment.  K compile-time => branch-free unrolled loop.
template<int K, int MT>
__global__ void k_pw_gemm_t(const float* __restrict__ W,   // (NEXP, M, K)
                            const float* __restrict__ X,   // per-sample or per-chunk-job
                            float* __restrict__ Y,         // (CHUNK, M, N)
                            const int* __restrict__ job_e, int job_base,
                            int M, int N, int x_per_sample)
{
    int cj = blockIdx.z;
    int gj = job_base + cj;
    int e  = job_e[gj];
    const float* Wt = W + (size_t)e * M * K;
    const float* Xb = X + (x_per_sample ? (size_t)(gj >> 1) * K * N
                                        : (size_t)cj * K * N);
    int m0 = blockIdx.y * (16 * MT);
    int n0 = blockIdx.x * 16;
    int lane = threadIdx.x & 31;
    int half = lane >> 4;     // K sub-pair select (ISA 16x4 f32 A/B layout)
    int l    = lane & 15;     // M index for A, N index for B
    int n    = n0 + l;

    v8f acc[MT];
#pragma unroll
    for (int t = 0; t < MT; ++t) acc[t] = (v8f){};

    constexpr int KP = (K + 3) & ~3;
#pragma unroll
    for (int k0 = 0; k0 < KP; k0 += 4) {
        int ka = k0 + half * 2;
        v2f bv;
        if constexpr ((K & 3) == 0) {
            bv.x = Xb[(size_t)ka * N + n];
            bv.y = Xb[(size_t)(ka + 1) * N + n];
        } else {
            // branch-free zero pad: clamp address (always in-bounds), mask value
            int   kx0 = (ka     < K) ? ka     : 0;
            int   kx1 = (ka + 1 < K) ? ka + 1 : 0;
            float mx0 = (ka     < K) ? 1.f : 0.f;
            float mx1 = (ka + 1 < K) ? 1.f : 0.f;
            bv.x = Xb[(size_t)kx0 * N + n] * mx0;
            bv.y = Xb[(size_t)kx1 * N + n] * mx1;
        }
#pragma unroll
        for (int t = 0; t < MT; ++t) {
            int m = m0 + t * 16 + l;
            v2f a;
            if constexpr ((K & 3) == 0) {
                a.x = Wt[(size_t)m * K + ka];        // contiguous pair -> b64 load
                a.y = Wt[(size_t)m * K + ka + 1];
            } else {
                int   kx0 = (ka     < K) ? ka     : 0;
                int   kx1 = (ka + 1 < K) ? ka + 1 : 0;
                float mx0 = (ka     < K) ? 1.f : 0.f;
                float mx1 = (ka + 1 < K) ? 1.f : 0.f;
                a.x = Wt[(size_t)m * K + kx0] * mx0;
                a.y = Wt[(size_t)m * K + kx1] * mx1;
            }
            acc[t] = __builtin_amdgcn_wmma_f32_16x16x4_f32(false, a, false, bv,
                                                           (short)0, acc[t], false, false);
        }
    }
    float* Yb = Y + (size_t)cj * M * N;
#pragma unroll
    for (int t = 0; t < MT; ++t)
#pragma unroll
        for (int r = 0; r < 8; ++r)
            Yb[(size_t)(m0 + t * 16 + half * 8 + r) * N + n] = acc[t][r];
}

// ---------------- GroupNorm stats (groups of 4 channels) -> per-channel affine --------
__global__ void k_gn_stats(const float* __restrict__ T,  // (CHUNK, C, HW)
                           const float* __restrict__ gs, const float* __restrict__ gb, // (NEXP, C)
                           const int* __restrict__ job_e, int job_base,
                           float* __restrict__ nA, float* __restrict__ nB,
                           int C, int HW)
{
    int cj = blockIdx.y;
    int g  = blockIdx.x;
    int e  = job_e[job_base + cj];
    const float* src = T + ((size_t)cj * C + (size_t)g * 4) * HW;
    int n = 4 * HW;
    float s = 0.f, ss = 0.f;
    for (int i = threadIdx.x; i < n; i += blockDim.x) {
        float v = src[i]; s += v; ss += v * v;
    }
    __shared__ float sh_s[256], sh_ss[256];
    sh_s[threadIdx.x] = s; sh_ss[threadIdx.x] = ss;
    __syncthreads();
    for (int off = 128; off > 0; off >>= 1) {
        if ((int)threadIdx.x < off) {
            sh_s[threadIdx.x]  += sh_s[threadIdx.x + off];
            sh_ss[threadIdx.x] += sh_ss[threadIdx.x + off];
        }
        __syncthreads();
    }
    if (threadIdx.x < 4) {
        float mean = sh_s[0] / n;
        float var  = sh_ss[0] / n - mean * mean;   // population var, eps 1e-5
        float rstd = rsqrtf(var + 1e-5f);
        int c = g * 4 + threadIdx.x;
        float A = gs[e * C + c] * rstd;
        nA[cj * MAXCH + c] = A;
        nB[cj * MAXCH + c] = gb[e * C + c] - mean * A;
    }
}

// ---------------- in-place affine (+ optional SiLU) -----------------------------------
__global__ void k_affine_act(float* __restrict__ T, const float* __restrict__ nA,
                             const float* __restrict__ nB, int C, int HW, int do_silu)
{
    int cj = blockIdx.y;
    size_t i = (size_t)blockIdx.x * blockDim.x + threadIdx.x;
    size_t tot = (size_t)C * HW;
    if (i >= tot) return;
    int c = (int)(i / HW);
    float* p = T + (size_t)cj * tot;
    float v = p[i] * nA[cj * MAXCH + c] + nB[cj * MAXCH + c];
    if (do_silu) v = v / (1.f + expf(-v));
    p[i] = v;
}

// ---------------- depthwise conv (zero-padded, strided) -------------------------------
__global__ void k_dwconv(const float* __restrict__ T1,
                         const float* __restrict__ W2,   // (NEXP, C, k, k)
                         const int* __restrict__ job_e, int job_base,
                         float* __restrict__ T2,
                         int C, int Hin, int Win, int Hout, int Wout, int ksz, int stride)
{
    int cj = blockIdx.z;
    int c  = blockIdx.y;
    int idx = blockIdx.x * blockDim.x + threadIdx.x;
    if (idx >= Hout * Wout) return;
    int oy = idx / Wout, ox = idx % Wout;
    int e  = job_e[job_base + cj];
    int pad = ksz / 2;
    const float* src = T1 + ((size_t)cj * C + c) * Hin * Win;
    const float* w   = W2 + ((size_t)e  * C + c) * ksz * ksz;
    float acc = 0.f;
    for (int ky = 0; ky < ksz; ++ky) {
        int iy = oy * stride - pad + ky;
        if (iy < 0 || iy >= Hin) continue;
        for (int kx = 0; kx < ksz; ++kx) {
            int ix = ox * stride - pad + kx;
            if (ix < 0 || ix >= Win) continue;
            acc += w[ky * ksz + kx] * src[iy * Win + ix];
        }
    }
    T2[((size_t)cj * C + c) * (Hout * Wout) + idx] = acc;
}

// ---------------- gate-weighted accumulation into next-stage input --------------------
__global__ void k_accum(const float* __restrict__ R3, const float* __restrict__ nA,
                        const float* __restrict__ nB, const float* __restrict__ jobs_g,
                        int job_base, float* __restrict__ Xnext, int C, int HW)
{
    int cj = blockIdx.y;
    int gj = job_base + cj;
    size_t i = (size_t)blockIdx.x * blockDim.x + threadIdx.x;
    size_t tot = (size_t)C * HW;
    if (i >= tot) return;
    int c = (int)(i / HW);
    float v = R3[(size_t)cj * tot + i] * nA[cj * MAXCH + c] + nB[cj * MAXCH + c];
    atomicAdd(&Xnext[(size_t)(gj >> 1) * tot + i], jobs_g[gj] * v);
}

__global__ void k_zero(float* __restrict__ p, size_t n)
{
    size_t i = (size_t)blockIdx.x * blockDim.x + threadIdx.x;
    if (i < n) p[i] = 0.f;
}

// ---------------- patchify (B,32,20,20) -> (B,25,512) + pos-embed, loss=0 -------------
__global__ void k_patchify(const float* __restrict__ X, const float* __restrict__ ppe,
                           float* __restrict__ out)
{
    const int TOT = BATCH * 25 * 512;
    int i = blockIdx.x * blockDim.x + threadIdx.x;
    if (i >= TOT) return;
    int d = i & 511;
    int p = (i >> 9) % 25;
    int b = i / (25 * 512);
    int c  = d & 31;
    int uv = d >> 5;
    int u = uv >> 2, v = uv & 3;
    int py = p / 5, px = p % 5;
    int h = py * 4 + u, w = px * 4 + v;
    out[i] = X[(((size_t)b * 32 + c) * 20 + h) * 20 + w] + ppe[p * 512 + d];
    if (i == 0) out[TOT] = 0.f;   // aux loss (all weights zero)
}

// ---------------- host-side GEMM dispatch (K, MT compile-time) ------------------------
static void launch_gemm(const float* W, const float* X, float* Y, const int* je,
                        int cb, int M, int K, int N, int xps, hipStream_t stream)
{
    int MT = (M >= 64) ? 4 : (M == 32 ? 2 : 1);
#define GEMM_CASE(KK, MM)                                                         \
    if (K == KK && MT == MM) {                                                    \
        dim3 grd(N / 16, M / (16 * MM), CHUNK);                                   \
        k_pw_gemm_t<KK, MM><<<grd, 32, 0, stream>>>(W, X, Y, je, cb, M, N, xps);  \
        return;                                                                   \
    }
    GEMM_CASE(3, 2)      // s1 pw1: 32x3
    GEMM_CASE(16, 4)     // s2 pw1: 64x16
    GEMM_CASE(32, 4)     // s3 pw1: 128x32
    GEMM_CASE(64, 4)     // s4/s5 pw1: 256x64 / 128x64, s3 pw3 uses 128 below
    GEMM_CASE(32, 1)     // s1 pw3: 16x32
    GEMM_CASE(64, 2)     // s2 pw3: 32x64
    GEMM_CASE(128, 4)    // s3 pw3: 64x128
    GEMM_CASE(256, 4)    // s4 pw3: 64x256
    GEMM_CASE(128, 2)    // s5 pw3: 32x128
#undef GEMM_CASE
}

extern "C" void kernel_launch(void* const* d_in, const int* in_sizes, int n_in,
                              void* d_out, int out_size, void* d_ws, size_t ws_size,
                              hipStream_t stream)
{
    (void)in_sizes; (void)n_in; (void)out_size; (void)ws_size;

    const float* gate_input   = (const float*)d_in[0];
    const float* expert_input = (const float*)d_in[1];
    const float* ppe          = (const float*)d_in[2];
    const int*   task         = (const int*)d_in[53];

    struct St { int cin, exp, cout, ksz, stride, Hin, Win, Hout, Wout; };
    const St st[5] = {
        { 3,  32, 16, 7, 2, 80, 80, 40, 40},
        {16,  64, 32, 3, 1, 40, 40, 40, 40},
        {32, 128, 64, 3, 2, 40, 40, 20, 20},
        {64, 256, 64, 3, 1, 20, 20, 20, 20},
        {64, 128, 32, 3, 1, 20, 20, 20, 20},
    };

    // ---- workspace layout (floats), ~72 MB total ----
    float* ws = (float*)d_ws;
    size_t off = 0;
    float* x_a = ws + off; off += (size_t)BATCH * 51200;     // max per-sample stage out (32*1600)
    float* x_b = ws + off; off += (size_t)BATCH * 51200;
    float* t1  = ws + off; off += (size_t)CHUNK * 204800;    // max exp*HWin (32*6400)
    float* t2  = ws + off; off += (size_t)CHUNK * 102400;    // max exp*HWout (256*400)
    float* r3  = ws + off; off += (size_t)CHUNK * 51200;     // max cout*HWout (32*1600)
    float* nA1 = ws + off; off += (size_t)CHUNK * MAXCH;
    float* nB1 = ws + off; off += (size_t)CHUNK * MAXCH;
    float* nA2 = ws + off; off += (size_t)CHUNK * MAXCH;
    float* nB2 = ws + off; off += (size_t)CHUNK * MAXCH;
    float* nA3 = ws + off; off += (size_t)CHUNK * MAXCH;
    float* nB3 = ws + off; off += (size_t)CHUNK * MAXCH;
    float* jobs_g = ws + off; off += (size_t)5 * JOBS;
    int*   jobs_e = (int*)(ws + off); off += (size_t)5 * JOBS;

    k_gating<<<1, 64, 0, stream>>>(gate_input,
        (const float*)d_in[3],  (const float*)d_in[13], (const float*)d_in[23],
        (const float*)d_in[33], (const float*)d_in[43],
        task, jobs_e, jobs_g);

    const float* xin  = expert_input;
    float* xout  = x_a;
    float* other = x_b;

    for (int s = 0; s < 5; ++s) {
        const St S = st[s];
        int base = 3 + s * 10;
        const float* w1  = (const float*)d_in[base + 1];
        const float* g1s = (const float*)d_in[base + 2];
        const float* g1b = (const float*)d_in[base + 3];
        const float* w2  = (const float*)d_in[base + 4];
        const float* g2s = (const float*)d_in[base + 5];
        const float* g2b = (const float*)d_in[base + 6];
        const float* w3  = (const float*)d_in[base + 7];
        const float* g3s = (const float*)d_in[base + 8];
        const float* g3b = (const float*)d_in[base + 9];
        int HWin  = S.Hin * S.Win;
        int HWout = S.Hout * S.Wout;
        const int*   je = jobs_e + s * JOBS;
        const float* jg = jobs_g + s * JOBS;

        size_t outN = (size_t)BATCH * S.cout * HWout;
        k_zero<<<dim3((unsigned)((outN + 255) / 256)), 256, 0, stream>>>(xout, outN);

        for (int cb = 0; cb < JOBS; cb += CHUNK) {
            // pw1 GEMM: (exp x cin) * (cin x HWin)
            launch_gemm(w1, xin, t1, je, cb, S.exp, S.cin, HWin, 1, stream);
            // GN1 -> affine, then SiLU in place
            k_gn_stats<<<dim3(S.exp / 4, CHUNK), 256, 0, stream>>>(
                t1, g1s, g1b, je, cb, nA1, nB1, S.exp, HWin);
            size_t n1 = (size_t)S.exp * HWin;
            k_affine_act<<<dim3((unsigned)((n1 + 255) / 256), CHUNK), 256, 0, stream>>>(
                t1, nA1, nB1, S.exp, HWin, 1);
            // depthwise conv
            k_dwconv<<<dim3((HWout + 127) / 128, S.exp, CHUNK), 128, 0, stream>>>(
                t1, w2, je, cb, t2, S.exp, S.Hin, S.Win, S.Hout, S.Wout, S.ksz, S.stride);
            // GN2 + SiLU
            k_gn_stats<<<dim3(S.exp / 4, CHUNK), 256, 0, stream>>>(
                t2, g2s, g2b, je, cb, nA2, nB2, S.exp, HWout);
            size_t n2 = (size_t)S.exp * HWout;
            k_affine_act<<<dim3((unsigned)((n2 + 255) / 256), CHUNK), 256, 0, stream>>>(
                t2, nA2, nB2, S.exp, HWout, 1);
            // pw3 GEMM: (cout x exp) * (exp x HWout)
            launch_gemm(w3, t2, r3, je, cb, S.cout, S.exp, HWout, 0, stream);
            // GN3 (no SiLU) folded into accumulation
            k_gn_stats<<<dim3(S.cout / 4, CHUNK), 256, 0, stream>>>(
                r3, g3s, g3b, je, cb, nA3, nB3, S.cout, HWout);
            size_t n3 = (size_t)S.cout * HWout;
            k_accum<<<dim3((unsigned)((n3 + 255) / 256), CHUNK), 256, 0, stream>>>(
                r3, nA3, nB3, jg, cb, xout, S.cout, HWout);
        }
        xin = xout;
        float* tmp = xout; xout = other; other = tmp;
    }

    // xin now points at the final stage output (B,32,20,20)
    k_patchify<<<(BATCH * 25 * 512 + 255) / 256, 256, 0, stream>>>(xin, ppe, (float*)d_out);
}